// EnhancedGCN_6777458393772
// MI455X (gfx1250) — compile-verified
//
#include <hip/hip_runtime.h>

typedef float v2f __attribute__((ext_vector_type(2)));
typedef float v8f __attribute__((ext_vector_type(8)));

// ---------------------------------------------------------------------------
// Degree accumulation over destination nodes (self-loop +1 folded into dinv).
// ---------------------------------------------------------------------------
__global__ void deg_kernel(const int* __restrict__ dst, unsigned* __restrict__ deg, int nE) {
    int t = blockIdx.x * blockDim.x + threadIdx.x;
    if (t < nE) atomicAdd(&deg[dst[t]], 1u);
}

__global__ void dinv_kernel(const unsigned* __restrict__ deg, float* __restrict__ dinv, int n) {
    int t = blockIdx.x * blockDim.x + threadIdx.x;
    if (t < n) dinv[t] = rsqrtf((float)deg[t] + 1.0f);   // deg incl. self-loop
}

// ---------------------------------------------------------------------------
// FP32 WMMA GEMM: C[M,N] = A[M,K] * B[K,N], row-major. One wave per 16x16
// output tile, N/16 waves per block, one M-tile per block. K stepped by 4
// with V_WMMA_F32_16X16X4_F32. M must be a multiple of 16 (100000 is).
// A fragment (16x4 f32): lanes 0-15 -> M=lane, VGPR0/1 = K={0,1}; lanes
// 16-31 -> K={2,3}. B fragment (4x16): mirrored, N striped across lanes.
// ---------------------------------------------------------------------------
template<int N>
__global__ __launch_bounds__(256) void gemm_wmma_f32(const float* __restrict__ A,
                                                     const float* __restrict__ B,
                                                     float* __restrict__ C, int K) {
    const int lane  = threadIdx.x & 31;
    const int wave  = threadIdx.x >> 5;      // 0 .. N/16-1
    const int mtile = blockIdx.x;
    const int lo    = lane & 15;
    const int hi    = lane >> 4;             // 0 or 1
    const int m     = mtile * 16 + lo;       // A row for this lane
    const int n     = wave * 16 + lo;        // C/B column for this lane

    const float* arow = A + (size_t)m * K;
    v8f acc = {};
    for (int k0 = 0; k0 < K; k0 += 4) {
        // A: contiguous K-pair per lane -> 8-byte vector load
        v2f a = *(const v2f*)(arow + k0 + 2 * hi);
        v2f b;
        b.x = B[(size_t)(k0 + 2 * hi)     * N + n];
        b.y = B[(size_t)(k0 + 2 * hi + 1) * N + n];
        acc = __builtin_amdgcn_wmma_f32_16x16x4_f32(
            /*neg_a=*/false, a, /*neg_b=*/false, b,
            /*c_mod=*/(short)0, acc, /*reuse_a=*/false, /*reuse_b=*/false);
    }
    // D layout: VGPR r, lanes 0-15 -> M=r, lanes 16-31 -> M=r+8; N = lane&15
#pragma unroll
    for (int r = 0; r < 8; ++r) {
        int row = mtile * 16 + r + 8 * hi;
        C[(size_t)row * N + n] = acc[r];
    }
}

// ---------------------------------------------------------------------------
// Edge-parallel normalized scatter-add: agg[dst] += h[src] * dinv[src]*dinv[dst]
// F = 4 << SHIFT features; lanes of a wave cover consecutive float4 chunks of
// one edge's feature row -> coalesced 512B gather + contiguous f32 atomics
// (L2-resident: 51.2MB accumulator << 192MB L2).
// ---------------------------------------------------------------------------
template<int SHIFT>
__global__ __launch_bounds__(256) void scatter_kernel(const float* __restrict__ h,
                                                      const int* __restrict__ src,
                                                      const int* __restrict__ dst,
                                                      const float* __restrict__ dinv,
                                                      float* __restrict__ agg, int nE) {
    const int F = 4 << SHIFT;
    long long t = (long long)blockIdx.x * blockDim.x + threadIdx.x;
    int e = (int)(t >> SHIFT);
    if (e >= nE) return;
    int c = ((int)t & ((1 << SHIFT) - 1)) * 4;
    int s = src[e];
    int d = dst[e];
    float nrm = dinv[s] * dinv[d];
    float4 v = *(const float4*)(h + (size_t)s * F + c);
    float* o = agg + (size_t)d * F + c;
    atomicAdd(o + 0, v.x * nrm);
    atomicAdd(o + 1, v.y * nrm);
    atomicAdd(o + 2, v.z * nrm);
    atomicAdd(o + 3, v.w * nrm);
}

// ---------------------------------------------------------------------------
// Finalize: out = act(agg + hlin * dinv^2 (self-loop) + bias)
// ---------------------------------------------------------------------------
template<int F, bool RELU>
__global__ __launch_bounds__(256) void finalize_kernel(const float* __restrict__ agg,
                                                       const float* __restrict__ hlin,
                                                       const float* __restrict__ dinv,
                                                       const float* __restrict__ bias,
                                                       float* __restrict__ out, int n) {
    const int CPN = F / 4;                   // float4 chunks per node
    int t = blockIdx.x * blockDim.x + threadIdx.x;
    if (t >= n * CPN) return;
    int node = t / CPN;
    int c = (t - node * CPN) * 4;
    float sn = dinv[node];
    sn = sn * sn;                            // self-loop norm dinv[i]^2
    float4 a = *(const float4*)(agg  + (size_t)node * F + c);
    float4 h = *(const float4*)(hlin + (size_t)node * F + c);
    float4 r;
    r.x = a.x + h.x * sn + bias[c + 0];
    r.y = a.y + h.y * sn + bias[c + 1];
    r.z = a.z + h.z * sn + bias[c + 2];
    r.w = a.w + h.w * sn + bias[c + 3];
    if (RELU) {
        r.x = fmaxf(r.x, 0.0f); r.y = fmaxf(r.y, 0.0f);
        r.z = fmaxf(r.z, 0.0f); r.w = fmaxf(r.w, 0.0f);
    }
    *(float4*)(out + (size_t)node * F + c) = r;
}

extern "C" void kernel_launch(void* const* d_in, const int* in_sizes, int n_in,
                              void* d_out, int out_size, void* d_ws, size_t ws_size,
                              hipStream_t stream) {
    const float* x  = (const float*)d_in[0];
    const int*   ei = (const int*)  d_in[1];
    const float* W1 = (const float*)d_in[2];
    const float* b1 = (const float*)d_in[3];
    const float* W2 = (const float*)d_in[4];
    const float* b2 = (const float*)d_in[5];
    const float* W3 = (const float*)d_in[6];
    const float* b3 = (const float*)d_in[7];

    const int IN = 128, HID = 128, OUT = 64;
    const int nN = in_sizes[0] / IN;         // 100000 (multiple of 16)
    const int nE = in_sizes[1] / 2;          // 1600000
    const int* src = ei;
    const int* dst = ei + nE;

    // workspace carving
    float*    hlin = (float*)d_ws;                       // [nN,128]
    float*    agg  = hlin + (size_t)nN * HID;            // [nN,128]
    float*    hact = agg  + (size_t)nN * HID;            // [nN,128]
    float*    dinv = hact + (size_t)nN * HID;            // [nN]
    unsigned* deg  = (unsigned*)(dinv + nN);             // [nN]

    // degree -> dinv
    hipMemsetAsync(deg, 0, (size_t)nN * sizeof(unsigned), stream);
    deg_kernel<<<(nE + 255) / 256, 256, 0, stream>>>(dst, deg, nE);
    dinv_kernel<<<(nN + 255) / 256, 256, 0, stream>>>(deg, dinv, nN);

    const int mtiles = nN / 16;

    // ---- layer 1: x @ W1 -> aggregate -> relu ----
    gemm_wmma_f32<128><<<mtiles, 256, 0, stream>>>(x, W1, hlin, IN);
    hipMemsetAsync(agg, 0, (size_t)nN * HID * sizeof(float), stream);
    scatter_kernel<5><<<(int)(((long long)nE * 32 + 255) / 256), 256, 0, stream>>>(
        hlin, src, dst, dinv, agg, nE);
    finalize_kernel<128, true><<<(nN * 32 + 255) / 256, 256, 0, stream>>>(
        agg, hlin, dinv, b1, hact, nN);

    // ---- layer 2 ----
    gemm_wmma_f32<128><<<mtiles, 256, 0, stream>>>(hact, W2, hlin, HID);
    hipMemsetAsync(agg, 0, (size_t)nN * HID * sizeof(float), stream);
    scatter_kernel<5><<<(int)(((long long)nE * 32 + 255) / 256), 256, 0, stream>>>(
        hlin, src, dst, dinv, agg, nE);
    finalize_kernel<128, true><<<(nN * 32 + 255) / 256, 256, 0, stream>>>(
        agg, hlin, dinv, b2, hact, nN);

    // ---- layer 3 (OUT=64, no relu) -> d_out ----
    gemm_wmma_f32<64><<<mtiles, 128, 0, stream>>>(hact, W3, hlin, HID);
    hipMemsetAsync(agg, 0, (size_t)nN * OUT * sizeof(float), stream);
    scatter_kernel<4><<<(int)(((long long)nE * 16 + 255) / 256), 256, 0, stream>>>(
        hlin, src, dst, dinv, agg, nE);
    finalize_kernel<64, false><<<(nN * 16 + 255) / 256, 256, 0, stream>>>(
        agg, hlin, dinv, b3, (float*)d_out, nN);
}